// GCN_214748365432
// MI455X (gfx1250) — compile-verified
//
#include <hip/hip_runtime.h>
#include <cstdint>

// ---------------------------------------------------------------------------
// CDNA5 (gfx1250) GCN-highway-GCN pipeline.
// GEMMs use V_WMMA_F32_16X16X4_F32 (wave32, 16x16 tile per wave).
// ---------------------------------------------------------------------------

typedef __attribute__((ext_vector_type(2))) float v2f;
typedef __attribute__((ext_vector_type(8))) float v8f;

__device__ __forceinline__ v8f wmma_f32_4(v2f a, v2f b, v8f c) {
  // 8 args: (neg_a, A, neg_b, B, c_mod, C, reuse_a, reuse_b)
  return __builtin_amdgcn_wmma_f32_16x16x4_f32(false, a, false, b, (short)0, c,
                                               false, false);
}

// ---------------------------- degree / dinv --------------------------------

__global__ void init_deg_kernel(float* deg, int n) {
  int i = blockIdx.x * blockDim.x + threadIdx.x;
  if (i < n) deg[i] = 1.0f;  // self loop
}

__global__ void count_deg_kernel(const int* __restrict__ col, float* deg, int e) {
  int i = blockIdx.x * blockDim.x + threadIdx.x;
  if (i < e) atomicAdd(&deg[col[i]], 1.0f);
}

__global__ void rsqrt_kernel(float* d, int n) {
  int i = blockIdx.x * blockDim.x + threadIdx.x;
  if (i < n) d[i] = __frsqrt_rn(d[i]);
}

// ------------------------------- GEMM --------------------------------------
// Y[nrows x NDIM] = X[nrows x 128] @ W[128 x NDIM] + bias.
// One block = 16-row strip; one wave32 per 16-column tile.

template <int NDIM>
__global__ void __launch_bounds__(32 * (NDIM / 16))
gemm_bias_wmma(const float* __restrict__ X, const float* __restrict__ W,
               const float* __restrict__ bias, float* __restrict__ Y, int nrows) {
  constexpr int K = 128;
  __shared__ float Xs[16][K + 4];  // pad: LDS bank = (4*m + k) % 64, conflict-free

  const int row0 = blockIdx.x * 16;
  const int tid  = threadIdx.x;
  const int lane = tid & 31;
  const int wave = tid >> 5;

  for (int i = tid; i < 16 * K; i += blockDim.x) {
    int r = i >> 7, c = i & (K - 1);
    int gr = row0 + r;
    Xs[r][c] = (gr < nrows) ? X[(size_t)gr * K + c] : 0.0f;
  }
  __syncthreads();

  const int ncol0 = wave * 16;
  const int m     = lane & 15;          // row (A) / col (B,C) within tile
  const int koff  = (lane >> 4) << 1;   // lanes 16-31 hold K+2,K+3

  v8f acc = {};
#pragma unroll
  for (int k0 = 0; k0 < K; k0 += 4) {
    v2f a, b;
    a.x = Xs[m][k0 + koff];
    a.y = Xs[m][k0 + koff + 1];
    b.x = W[(size_t)(k0 + koff) * NDIM + ncol0 + m];
    b.y = W[(size_t)(k0 + koff + 1) * NDIM + ncol0 + m];
    acc = wmma_f32_4(a, b, acc);
  }

  const float bv = bias[ncol0 + m];
#pragma unroll
  for (int r = 0; r < 8; ++r) {
    int mm = r + ((lane >> 4) << 3);
    int gr = row0 + mm;
    if (gr < nrows) Y[(size_t)gr * NDIM + ncol0 + m] = acc[r] + bv;
  }
}

// --------------------------- aggregation -----------------------------------
// out[n] = dinv[n]^2 * hlin[n]  (self-loop term), then edge scatter-add.

template <int F>
__global__ void agg_init_kernel(const float* __restrict__ hlin,
                                const float* __restrict__ dinv,
                                float* __restrict__ out, int n) {
  int i = blockIdx.x * blockDim.x + threadIdx.x;
  if (i < n * F) {
    float d = dinv[i / F];
    out[i] = d * d * hlin[i];
  }
}

template <int F>
__global__ void __launch_bounds__(256)
agg_edges_kernel(const int* __restrict__ rows, const int* __restrict__ cols,
                 const float* __restrict__ dinv, const float* __restrict__ hlin,
                 float* __restrict__ out, int ne) {
  int e = blockIdx.x * (blockDim.x >> 5) + (threadIdx.x >> 5);
  if (e >= ne) return;
  const int lane = threadIdx.x & 31;
  const int r = rows[e];
  const int c = cols[e];
  const float nrm = dinv[r] * dinv[c];
  const float* __restrict__ src = hlin + (size_t)r * F;
  float* __restrict__ dst = out + (size_t)c * F;
#pragma unroll
  for (int f = lane; f < F; f += 32) atomicAdd(&dst[f], nrm * src[f]);
}

// ------------------------------ highway ------------------------------------
// out = g * relu(x@Wn+bn) + (1-g) * x,  g = softmax(x@Wg+bg, axis=1).
// One block = 16-row strip; each of 8 waves computes both the "normal" and
// "gate" 16x16 tiles for its column range (two WMMA accumulator chains).

__global__ void __launch_bounds__(256)
highway_wmma(const float* __restrict__ x, const float* __restrict__ Wn,
             const float* __restrict__ bn, const float* __restrict__ Wg,
             const float* __restrict__ bg, float* __restrict__ out, int nrows) {
  constexpr int K = 128;
  __shared__ float Xs[16][K + 4];
  __shared__ float Ns[16][K + 4];
  __shared__ float Gs[16][K + 4];

  const int row0 = blockIdx.x * 16;
  const int tid  = threadIdx.x;
  const int lane = tid & 31;
  const int wave = tid >> 5;

  for (int i = tid; i < 16 * K; i += 256) {
    int r = i >> 7, c = i & (K - 1);
    int gr = row0 + r;
    Xs[r][c] = (gr < nrows) ? x[(size_t)gr * K + c] : 0.0f;
  }
  __syncthreads();

  const int ncol0 = wave * 16;
  const int m     = lane & 15;
  const int koff  = (lane >> 4) << 1;

  v8f accN = {};
  v8f accG = {};
#pragma unroll
  for (int k0 = 0; k0 < K; k0 += 4) {
    v2f a, b0, b1;
    a.x  = Xs[m][k0 + koff];
    a.y  = Xs[m][k0 + koff + 1];
    b0.x = Wn[(size_t)(k0 + koff) * K + ncol0 + m];
    b0.y = Wn[(size_t)(k0 + koff + 1) * K + ncol0 + m];
    b1.x = Wg[(size_t)(k0 + koff) * K + ncol0 + m];
    b1.y = Wg[(size_t)(k0 + koff + 1) * K + ncol0 + m];
    accN = wmma_f32_4(a, b0, accN);
    accG = wmma_f32_4(a, b1, accG);
  }

  const float bnv = bn[ncol0 + m];
  const float bgv = bg[ncol0 + m];
#pragma unroll
  for (int r = 0; r < 8; ++r) {
    int mm = r + ((lane >> 4) << 3);
    Ns[mm][ncol0 + m] = fmaxf(accN[r] + bnv, 0.0f);
    Gs[mm][ncol0 + m] = accG[r] + bgv;
  }
  __syncthreads();

  // row-wise softmax over 128 gate logits (one thread per row)
  if (tid < 16) {
    float mx = -3.402823466e+38f;
    for (int c = 0; c < K; ++c) mx = fmaxf(mx, Gs[tid][c]);
    float s = 0.0f;
    for (int c = 0; c < K; ++c) {
      float ev = __expf(Gs[tid][c] - mx);
      Gs[tid][c] = ev;
      s += ev;
    }
    float inv = 1.0f / s;
    for (int c = 0; c < K; ++c) Gs[tid][c] *= inv;
  }
  __syncthreads();

  for (int i = tid; i < 16 * K; i += 256) {
    int r = i >> 7, c = i & (K - 1);
    int gr = row0 + r;
    if (gr < nrows) {
      float g = Gs[r][c];
      out[(size_t)gr * K + c] = g * Ns[r][c] + (1.0f - g) * Xs[r][c];
    }
  }
}

// ------------------------------ launcher -----------------------------------

extern "C" void kernel_launch(void* const* d_in, const int* in_sizes, int n_in,
                              void* d_out, int out_size, void* d_ws, size_t ws_size,
                              hipStream_t stream) {
  constexpr int IN = 128, H = 128, C = 64;

  const float* x    = (const float*)d_in[0];
  const int*   ei   = (const int*)d_in[1];
  const float* W1   = (const float*)d_in[2];
  const float* b1   = (const float*)d_in[3];
  const float* W2   = (const float*)d_in[4];
  const float* b2   = (const float*)d_in[5];
  const float* Wn   = (const float*)d_in[6];
  const float* bn   = (const float*)d_in[7];
  const float* Wg   = (const float*)d_in[8];
  const float* bg   = (const float*)d_in[9];
  float*       out  = (float*)d_out;

  const int N = in_sizes[0] / IN;
  const int E = in_sizes[1] / 2;
  const int* erow = ei;       // source nodes
  const int* ecol = ei + E;   // destination nodes

  // workspace layout
  float* dinv = (float*)d_ws;
  float* bufA = dinv + (((size_t)N + 1023) & ~(size_t)1023);
  float* bufB = bufA + (size_t)N * H;
  (void)ws_size; (void)n_in; (void)out_size;

  const int T = 256;
  const int rowBlocks = (N + 15) / 16;

  // 1) degrees -> dinv
  init_deg_kernel<<<(N + T - 1) / T, T, 0, stream>>>(dinv, N);
  count_deg_kernel<<<(E + T - 1) / T, T, 0, stream>>>(ecol, dinv, E);
  rsqrt_kernel<<<(N + T - 1) / T, T, 0, stream>>>(dinv, N);

  // 2) h_lin1 = x @ W1 + b1            (bufA)
  gemm_bias_wmma<H><<<rowBlocks, 256, 0, stream>>>(x, W1, b1, bufA, N);

  // 3) h1 = normalized aggregate(h_lin1)   (bufB)
  agg_init_kernel<H><<<((size_t)N * H + T - 1) / T, T, 0, stream>>>(bufA, dinv, bufB, N);
  agg_edges_kernel<H><<<(E + 7) / 8, 256, 0, stream>>>(erow, ecol, dinv, bufA, bufB, E);

  // 4) hw = highway(h1)                (bufA)
  highway_wmma<<<rowBlocks, 256, 0, stream>>>(bufB, Wn, bn, Wg, bg, bufA, N);

  // 5) h_lin2 = hw @ W2 + b2           (bufB, first N*C floats)
  gemm_bias_wmma<C><<<rowBlocks, 128, 0, stream>>>(bufA, W2, b2, bufB, N);

  // 6) out = normalized aggregate(h_lin2)
  agg_init_kernel<C><<<((size_t)N * C + T - 1) / T, T, 0, stream>>>(bufB, dinv, out, N);
  agg_edges_kernel<C><<<(E + 7) / 8, 256, 0, stream>>>(erow, ecol, dinv, bufB, out, E);
}